// AdaptiveMaskGenerator_8392366096633
// MI455X (gfx1250) — compile-verified
//
#include <hip/hip_runtime.h>
#include <hip/hip_bf16.h>
#include <stdint.h>

// Reference: x [16,307,64,288] f32 -> per-row top-72 by |x| -> 0/1 mask (f32).
// Row-wise k-selection; memory-bound (~724 MB total traffic @ 23.3 TB/s ~ 31 us).

constexpr int kL            = 288;        // time length
constexpr int kK            = 72;         // num_mask = int(288 * 0.25)
constexpr int kWave         = 32;         // CDNA5 is wave32-only
constexpr int kJ            = kL / kWave; // 9 keys per lane
constexpr int kRowsPerBlock = 8;          // 8 waves per 256-thread block

// gfx1250 async global->LDS copy path (ASYNCcnt), guarded for portability.
#if defined(__HIP_DEVICE_COMPILE__) && __has_builtin(__builtin_amdgcn_global_load_async_to_lds_b128)
#define MASKGEN_ASYNC 1
#else
#define MASKGEN_ASYNC 0
#endif

#if MASKGEN_ASYNC
// Builtin signature (from hipcc diagnostic): first arg is AS1 pointer to
// int __attribute__((vector_size(16))), second is the AS3 (LDS) counterpart.
typedef int v4i __attribute__((vector_size(4 * sizeof(int))));
typedef __attribute__((address_space(1))) v4i* g_v4i_ptr;
typedef __attribute__((address_space(3))) v4i* l_v4i_ptr;
#endif

__device__ __forceinline__ unsigned ballot32(bool p) {
#if defined(__HIP_DEVICE_COMPILE__)
  return __builtin_amdgcn_ballot_w32(p);
#else
  (void)p; return 0u;
#endif
}

__global__ __launch_bounds__(256)
void adaptive_mask_topk_kernel(const float* __restrict__ x,
                               float* __restrict__ out,
                               int num_rows) {
  const int lane = threadIdx.x & (kWave - 1);
  const int wv   = threadIdx.x >> 5;
  const int row  = blockIdx.x * kRowsPerBlock + wv;
  if (row >= num_rows) return;                 // wave-uniform condition
  const size_t base = (size_t)row * kL;

#if MASKGEN_ASYNC
  __shared__ alignas(16) unsigned lds[kRowsPerBlock][kL];
  {
    // 288 floats = 72 x b128 chunks, striped over 32 lanes: 32 + 32 + 8.
    const char* g = (const char*)(x + base);
    char*       l = (char*)&lds[wv][0];
    __builtin_amdgcn_global_load_async_to_lds_b128(
        (g_v4i_ptr)(g + (size_t)lane * 16),
        (l_v4i_ptr)(l + lane * 16), 0, 0);
    __builtin_amdgcn_global_load_async_to_lds_b128(
        (g_v4i_ptr)(g + (size_t)(lane + 32) * 16),
        (l_v4i_ptr)(l + (lane + 32) * 16), 0, 0);
    if (lane < 8) {
      __builtin_amdgcn_global_load_async_to_lds_b128(
          (g_v4i_ptr)(g + (size_t)(lane + 64) * 16),
          (l_v4i_ptr)(l + (lane + 64) * 16), 0, 0);
    }
#if __has_builtin(__builtin_amdgcn_s_wait_asynccnt)
    __builtin_amdgcn_s_wait_asynccnt(0);
#else
    asm volatile("s_wait_asynccnt 0" ::: "memory");
#endif
  }
#endif

  // |x| as unsigned key: clears sign bit; uint order == |float| order.
  unsigned key[kJ];
#pragma unroll
  for (int j = 0; j < kJ; ++j) {
    const int idx = lane + j * kWave;          // coalesced / conflict-free layout
#if MASKGEN_ASYNC
    key[j] = lds[wv][idx] & 0x7fffffffu;
#else
    key[j] = __float_as_uint(x[base + idx]) & 0x7fffffffu;
#endif
  }

  // count(#keys >= t) across the wave: 9x (v_cmp -> ballot -> s_bcnt),
  // result is wave-uniform (lives in SGPRs).
  auto count_ge = [&](unsigned t) -> int {
    int c = 0;
#pragma unroll
    for (int j = 0; j < kJ; ++j)
      c += __builtin_popcount(ballot32(key[j] >= t));
    return c;
  };

  // Uniform binary search for the largest T with count_ge(T) >= 72.
  // Keys live in [0, 0x7f800000): ~31 iterations, fully wave-uniform.
  unsigned lo = 0u, hi = 0x7f7fffffu;
  while (lo < hi) {
    const unsigned mid = lo + ((hi - lo + 1u) >> 1);
    if (count_ge(mid) >= kK) lo = mid; else hi = mid - 1u;
  }
  const unsigned thr = lo;

  // Exact-k tie handling (top_k breaks ties toward lower index):
  // mark all strictly-greater, then the first (kK - cnt_gt) ties in index order.
  const int need = kK - count_ge(thr + 1u);
  const unsigned below_mask = (1u << lane) - 1u;  // lanes strictly below me
  int running = 0;
#pragma unroll
  for (int j = 0; j < kJ; ++j) {
    const bool gt  = key[j] > thr;
    const bool tie = key[j] == thr;
    const unsigned tm = ballot32(tie);
    const int rank = running + __builtin_popcount(tm & below_mask);
    const bool pick = gt | (tie & (rank < need));
    out[base + lane + j * kWave] = pick ? 1.0f : 0.0f;   // coalesced b32 stores
    running += __builtin_popcount(tm);
  }
}

extern "C" void kernel_launch(void* const* d_in, const int* in_sizes, int n_in,
                              void* d_out, int out_size, void* d_ws, size_t ws_size,
                              hipStream_t stream) {
  (void)n_in; (void)out_size; (void)d_ws; (void)ws_size;
  const float* x   = (const float*)d_in[0];
  float*       out = (float*)d_out;

  const int total    = in_sizes[0];     // 16*307*64*288
  const int num_rows = total / kL;      // 314368

  dim3 block(256);
  dim3 grid((num_rows + kRowsPerBlock - 1) / kRowsPerBlock);
  hipLaunchKernelGGL(adaptive_mask_topk_kernel, grid, block, 0, stream,
                     x, out, num_rows);
}